// MiniMaxM2MoE_43233140801846
// MI455X (gfx1250) — compile-verified
//
#include <hip/hip_runtime.h>
#include <cstdint>

#define TT  2048
#define HH  2048
#define IIN 1024
#define EE  16
#define TOPK 4

typedef __attribute__((ext_vector_type(2))) float v2f;
typedef __attribute__((ext_vector_type(4))) float v4f;
typedef __attribute__((ext_vector_type(8))) float v8f;

__device__ inline v8f v8zero() {
  v8f z;
#pragma unroll
  for (int i = 0; i < 8; ++i) z[i] = 0.f;
  return z;
}

// Async copy 16B/lane straight from global memory into LDS (no VGPR bounce).
// Tracked by ASYNCcnt. LDS destination address = low 32 bits of the generic
// shared-memory pointer (LDS aperture encodes offset in addr[31:0]).
__device__ inline void async_ld128(void* lds_ptr, const void* gptr) {
  uint32_t l = (uint32_t)(uintptr_t)lds_ptr;
  asm volatile("global_load_async_to_lds_b128 %0, %1, off"
               :
               : "v"(l), "v"(gptr)
               : "memory");
}

__device__ inline void wait_async0() {
  asm volatile("s_wait_asynccnt 0x0" ::: "memory");
}

// ---------------------------------------------------------------------------
// Router: logits = X @ gate_w^T ; softmax over 16 experts; top-4; renormalize.
// One block per token. Produces dense combine weights cw[T, 16].
// ---------------------------------------------------------------------------
__global__ __launch_bounds__(256) void moe_router(const float* __restrict__ X,
                                                  const float* __restrict__ GW,
                                                  float* __restrict__ cw) {
  __shared__ float xs[HH];
  __shared__ float partial[EE][17];
  __shared__ float logits[EE];
  const int t = blockIdx.x;
  const int tid = threadIdx.x;
  for (int i = tid; i < HH; i += 256) xs[i] = X[(size_t)t * HH + i];
  __syncthreads();
  const int e = tid >> 4;
  const int s = tid & 15;
  float p = 0.f;
  const float* gw = GW + (size_t)e * HH;
  for (int h = s; h < HH; h += 16) p += xs[h] * gw[h];
  partial[e][s] = p;
  __syncthreads();
  if (s == 0) {
    float acc = 0.f;
#pragma unroll
    for (int i = 0; i < 16; ++i) acc += partial[e][i];
    logits[e] = acc;
  }
  __syncthreads();
  if (tid == 0) {
    float lg[EE];
    float mx = -1e30f;
#pragma unroll
    for (int i = 0; i < EE; ++i) { lg[i] = logits[i]; mx = fmaxf(mx, lg[i]); }
    float sum = 0.f;
#pragma unroll
    for (int i = 0; i < EE; ++i) { lg[i] = __expf(lg[i] - mx); sum += lg[i]; }
    const float inv = 1.f / sum;
#pragma unroll
    for (int i = 0; i < EE; ++i) lg[i] *= inv;
    float row[EE];
#pragma unroll
    for (int i = 0; i < EE; ++i) row[i] = 0.f;
    bool used[EE];
#pragma unroll
    for (int i = 0; i < EE; ++i) used[i] = false;
    int pick[TOPK];
    float pw[TOPK];
    float wsum = 0.f;
    for (int k = 0; k < TOPK; ++k) {
      int best = 0;
      float bv = -1.f;
      for (int i = 0; i < EE; ++i)
        if (!used[i] && lg[i] > bv) { bv = lg[i]; best = i; }
      used[best] = true;
      pick[k] = best;
      pw[k] = bv;
      wsum += bv;
    }
    const float winv = 1.f / wsum;
    for (int k = 0; k < TOPK; ++k) row[pick[k]] = pw[k] * winv;
#pragma unroll
    for (int i = 0; i < EE; ++i) cw[(size_t)t * EE + i] = row[i];
  }
}

// ---------------------------------------------------------------------------
// GEMM1 (per expert): act[t, i] = cw[t,e] * silu(X@w1)[t,i] * (X@w3)[t,i]
// Tile: 128 tokens x 64 I-cols; 8 waves in 4(M)x2(N); each wave owns 2x2
// 16x16 C-tiles for BOTH w1 and w3 streams (shared A frags).
// Double-buffered LDS fed by GLOBAL_LOAD_ASYNC_TO_LDS_B128 (ASYNCcnt),
// overlapping stage k+1 copies with the 64-WMMA stage-k compute block.
// ---------------------------------------------------------------------------
__global__ __launch_bounds__(256) void moe_gemm1(const float* __restrict__ X,
                                                 const float* __restrict__ W1,
                                                 const float* __restrict__ W3,
                                                 const float* __restrict__ cw,
                                                 float* __restrict__ act,
                                                 int e) {
  __shared__ float Xs[2][128 * 36];   // 128 rows x 32 K (+4 pad)
  __shared__ float W1s[2][32 * 68];   // 32 K x 64 N (+4 pad)
  __shared__ float W3s[2][32 * 68];
  __shared__ float cws[128];

  const int tid = threadIdx.x;
  const int lane = tid & 31;
  const int wave = tid >> 5;
  const int lm = lane & 15;
  const int khi = (lane >> 4) << 1;  // 0 | 2 : K-pair owned by lane half
  const int rhi = (lane >> 4) << 3;  // 0 | 8 : C-row offset of lane half
  const int wM = (wave & 3) * 32;
  const int wN = (wave >> 2) * 32;
  const int t0 = blockIdx.y * 128;
  const int i0 = blockIdx.x * 64;

  if (tid < 128) cws[tid] = cw[(size_t)(t0 + tid) * EE + e];

  auto stage = [&](int buf, int k0) {
#pragma unroll
    for (int j = 0; j < 4; ++j) {
      const int id = tid + j * 256;
      const int r = id >> 3;
      const int c = (id & 7) << 2;
      async_ld128(&Xs[buf][r * 36 + c], X + (size_t)(t0 + r) * HH + k0 + c);
    }
#pragma unroll
    for (int j = 0; j < 2; ++j) {
      const int id = tid + j * 256;
      const int r = id >> 4;
      const int c = (id & 15) << 2;
      async_ld128(&W1s[buf][r * 68 + c], W1 + (size_t)(k0 + r) * IIN + i0 + c);
      async_ld128(&W3s[buf][r * 68 + c], W3 + (size_t)(k0 + r) * IIN + i0 + c);
    }
  };

  v8f a1[2][2], a3[2][2];
#pragma unroll
  for (int m = 0; m < 2; ++m)
#pragma unroll
    for (int n = 0; n < 2; ++n) { a1[m][n] = v8zero(); a3[m][n] = v8zero(); }

  stage(0, 0);
  wait_async0();
  __syncthreads();

  int buf = 0;
  for (int k0 = 0; k0 < HH; k0 += 32, buf ^= 1) {
    if (k0 + 32 < HH) stage(buf ^ 1, k0 + 32);  // overlap with compute below
#pragma unroll
    for (int kk = 0; kk < 8; ++kk) {
      const int k4 = kk * 4 + khi;
      v2f a[2], b1[2], b3[2];
#pragma unroll
      for (int m = 0; m < 2; ++m) {
        const float* p = &Xs[buf][(wM + m * 16 + lm) * 36 + k4];
        a[m].x = p[0];
        a[m].y = p[1];
      }
#pragma unroll
      for (int n = 0; n < 2; ++n) {
        const float* p1 = &W1s[buf][k4 * 68 + wN + n * 16 + lm];
        b1[n].x = p1[0];
        b1[n].y = p1[68];
        const float* p3 = &W3s[buf][k4 * 68 + wN + n * 16 + lm];
        b3[n].x = p3[0];
        b3[n].y = p3[68];
      }
#pragma unroll
      for (int m = 0; m < 2; ++m)
#pragma unroll
        for (int n = 0; n < 2; ++n) {
          a1[m][n] = __builtin_amdgcn_wmma_f32_16x16x4_f32(
              false, a[m], false, b1[n], (short)0, a1[m][n], false, false);
          a3[m][n] = __builtin_amdgcn_wmma_f32_16x16x4_f32(
              false, a[m], false, b3[n], (short)0, a3[m][n], false, false);
        }
    }
    wait_async0();   // next stage landed in LDS
    __syncthreads(); // all waves done reading buf before it is rewritten
  }

  // Epilogue: SwiGLU + combine-weight scaling, store activation tile.
#pragma unroll
  for (int m = 0; m < 2; ++m)
#pragma unroll
    for (int n = 0; n < 2; ++n)
#pragma unroll
      for (int r = 0; r < 8; ++r) {
        const int rl = wM + m * 16 + rhi + r;   // local token row 0..127
        const int col = i0 + wN + n * 16 + lm;  // global I column
        const float h1 = a1[m][n][r];
        const float h3 = a3[m][n][r];
        const float sv = h1 / (1.f + __expf(-h1));  // silu
        act[(size_t)(t0 + rl) * IIN + col] = sv * h3 * cws[rl];
      }
}

// ---------------------------------------------------------------------------
// GEMM2 (per expert): out[t, h] (+)= act @ w2. first==1 overwrites (expert 0).
// Expert-serial launches on the stream -> deterministic accumulation.
// Same async double-buffered pipeline as GEMM1.
// ---------------------------------------------------------------------------
__global__ __launch_bounds__(256) void moe_gemm2(const float* __restrict__ A,
                                                 const float* __restrict__ W2,
                                                 float* __restrict__ out,
                                                 int first) {
  __shared__ float As[2][128 * 36];
  __shared__ float Ws[2][32 * 68];

  const int tid = threadIdx.x;
  const int lane = tid & 31;
  const int wave = tid >> 5;
  const int lm = lane & 15;
  const int khi = (lane >> 4) << 1;
  const int rhi = (lane >> 4) << 3;
  const int wM = (wave & 3) * 32;
  const int wN = (wave >> 2) * 32;
  const int t0 = blockIdx.y * 128;
  const int h0 = blockIdx.x * 64;

  auto stage = [&](int buf, int k0) {
#pragma unroll
    for (int j = 0; j < 4; ++j) {
      const int id = tid + j * 256;
      const int r = id >> 3;
      const int c = (id & 7) << 2;
      async_ld128(&As[buf][r * 36 + c], A + (size_t)(t0 + r) * IIN + k0 + c);
    }
#pragma unroll
    for (int j = 0; j < 2; ++j) {
      const int id = tid + j * 256;
      const int r = id >> 4;
      const int c = (id & 15) << 2;
      async_ld128(&Ws[buf][r * 68 + c], W2 + (size_t)(k0 + r) * HH + h0 + c);
    }
  };

  v8f acc[2][2];
#pragma unroll
  for (int m = 0; m < 2; ++m)
#pragma unroll
    for (int n = 0; n < 2; ++n) acc[m][n] = v8zero();

  stage(0, 0);
  wait_async0();
  __syncthreads();

  int buf = 0;
  for (int k0 = 0; k0 < IIN; k0 += 32, buf ^= 1) {
    if (k0 + 32 < IIN) stage(buf ^ 1, k0 + 32);
#pragma unroll
    for (int kk = 0; kk < 8; ++kk) {
      const int k4 = kk * 4 + khi;
      v2f a[2], b[2];
#pragma unroll
      for (int m = 0; m < 2; ++m) {
        const float* p = &As[buf][(wM + m * 16 + lm) * 36 + k4];
        a[m].x = p[0];
        a[m].y = p[1];
      }
#pragma unroll
      for (int n = 0; n < 2; ++n) {
        const float* p = &Ws[buf][k4 * 68 + wN + n * 16 + lm];
        b[n].x = p[0];
        b[n].y = p[68];
      }
#pragma unroll
      for (int m = 0; m < 2; ++m)
#pragma unroll
        for (int n = 0; n < 2; ++n)
          acc[m][n] = __builtin_amdgcn_wmma_f32_16x16x4_f32(
              false, a[m], false, b[n], (short)0, acc[m][n], false, false);
    }
    wait_async0();
    __syncthreads();
  }

#pragma unroll
  for (int m = 0; m < 2; ++m)
#pragma unroll
    for (int n = 0; n < 2; ++n)
#pragma unroll
      for (int r = 0; r < 8; ++r) {
        const int rl = wM + m * 16 + rhi + r;
        const int col = h0 + wN + n * 16 + lm;
        const size_t o = (size_t)(t0 + rl) * HH + col;
        const float v = acc[m][n][r];
        if (first) out[o] = v;
        else out[o] += v;
      }
}

// ---------------------------------------------------------------------------
extern "C" void kernel_launch(void* const* d_in, const int* in_sizes, int n_in,
                              void* d_out, int out_size, void* d_ws, size_t ws_size,
                              hipStream_t stream) {
  (void)in_sizes; (void)n_in; (void)out_size; (void)ws_size;
  const float* X  = (const float*)d_in[0];  // [T, H]
  const float* GW = (const float*)d_in[1];  // [E, H]
  const float* W1 = (const float*)d_in[2];  // [E, H, I]
  const float* W3 = (const float*)d_in[3];  // [E, H, I]
  const float* W2 = (const float*)d_in[4];  // [E, I, H]
  float* out = (float*)d_out;               // [T, H]

  float* cw  = (float*)d_ws;                // [T, E]      (128 KB)
  float* act = cw + (size_t)TT * EE;        // [T, I]      (8 MB)

  moe_router<<<TT, 256, 0, stream>>>(X, GW, cw);

  for (int e = 0; e < EE; ++e) {
    moe_gemm1<<<dim3(IIN / 64, TT / 128), 256, 0, stream>>>(
        X, W1 + (size_t)e * HH * IIN, W3 + (size_t)e * HH * IIN, cw, act, e);
    moe_gemm2<<<dim3(HH / 64, TT / 128), 256, 0, stream>>>(
        act, W2 + (size_t)e * IIN * HH, out, e == 0);
  }
}